// ImagePrompt_46050639348091
// MI455X (gfx1250) — compile-verified
//
#include <hip/hip_runtime.h>
#include <math.h>

// ---- problem constants (from reference) ------------------------------------
#define B_    256
#define N_    196
#define D_    768
#define P_    32
#define TOPK  5
#define LSEQ  216          // top_k*4 + N = 20 + 196
#define MROWS 128          // P_ * 4 patches

typedef __attribute__((ext_vector_type(2))) float v2f;
typedef __attribute__((ext_vector_type(8))) float v8f;

// ============================================================================
// Kernel 1: fused  out[b,20:216,:] = x_embed[b]  +  column mean over N
//           + L2-normalize the mean  ->  xnorm (B,768)
// One block per batch row; 192 threads x float4 = 768 cols, b128 coalesced.
// ============================================================================
__global__ void mean_norm_copy_kernel(const float* __restrict__ x,
                                      float* __restrict__ out,     // (B,216,768)
                                      float* __restrict__ xnorm) { // (B,768)
  const int b   = blockIdx.x;
  const int t   = threadIdx.x;          // 0..191
  const int col = t * 4;
  const float* xb = x + (size_t)b * N_ * D_;
  float* ob = out + (size_t)b * LSEQ * D_ + (size_t)(TOPK * 4) * D_;
  float4 acc = make_float4(0.f, 0.f, 0.f, 0.f);
  for (int n = 0; n < N_; ++n) {
    const float4 v = *(const float4*)(xb + (size_t)n * D_ + col);
    *(float4*)(ob + (size_t)n * D_ + col) = v;
    acc.x += v.x; acc.y += v.y; acc.z += v.z; acc.w += v.w;
  }
  const float inv = 1.0f / (float)N_;
  acc.x *= inv; acc.y *= inv; acc.z *= inv; acc.w *= inv;

  __shared__ float red[192];
  red[t] = acc.x * acc.x + acc.y * acc.y + acc.z * acc.z + acc.w * acc.w;
  __syncthreads();
  for (int s = 96; s >= 3; s >>= 1) {      // 192 = 3 * 2^6
    if (t < s) red[t] += red[t + s];
    __syncthreads();
  }
  if (t == 0) red[0] = red[0] + red[1] + red[2];
  __syncthreads();
  const float scale = rsqrtf(fmaxf(red[0], 1e-12f));
  float4 o;
  o.x = acc.x * scale; o.y = acc.y * scale; o.z = acc.z * scale; o.w = acc.w * scale;
  *(float4*)(xnorm + (size_t)b * D_ + col) = o;
}

// ============================================================================
// Kernel 2: L2-normalize rows of prompt_key -> pnorm (P,768)
// ============================================================================
__global__ void rownorm_kernel(const float* __restrict__ in,
                               float* __restrict__ outp) {
  const int r = blockIdx.x;
  const int t = threadIdx.x;            // 192
  const float4 v = *(const float4*)(in + (size_t)r * D_ + t * 4);
  __shared__ float red[192];
  red[t] = v.x * v.x + v.y * v.y + v.z * v.z + v.w * v.w;
  __syncthreads();
  for (int s = 96; s >= 3; s >>= 1) {
    if (t < s) red[t] += red[t + s];
    __syncthreads();
  }
  if (t == 0) red[0] = red[0] + red[1] + red[2];
  __syncthreads();
  const float scale = rsqrtf(fmaxf(red[0], 1e-12f));
  float4 o;
  o.x = v.x * scale; o.y = v.y * scale; o.z = v.z * scale; o.w = v.w * scale;
  *(float4*)(outp + (size_t)r * D_ + t * 4) = o;
}

// ============================================================================
// Kernel 3: patchify prompt (32,3,32,32) -> A (128,768), row = p*4 + l,
// l -> (pr,pc) = (l>>1, l&1); col = c*256 + i*16 + j (matches conv_w layout).
// ============================================================================
__global__ void patchify_kernel(const float* __restrict__ prompt,
                                float* __restrict__ A) {
  const int row = blockIdx.x;               // 0..127
  const int p = row >> 2, l = row & 3;
  const int pr = (l >> 1) * 16, pc = (l & 1) * 16;
  for (int e = threadIdx.x; e < D_; e += blockDim.x) {
    const int c = e >> 8, rem = e & 255, i = rem >> 4, j = rem & 15;
    A[(size_t)row * D_ + e] =
        prompt[(size_t)(((p * 3 + c) * 32) + pr + i) * 32 + pc + j];
  }
}

// ============================================================================
// Kernel 4: f32 WMMA GEMM  D = A(MxK) * Bt(NxK)^T + bias, one wave per
// 16x16 tile, K stepped by 4 through V_WMMA_F32_16X16X4_F32.
// A-frag (16x4, 32b): lane = half*16 + m16; VGPR0/1 = contiguous {k0+2h, k0+2h+1}
// of row m16.  B-frag symmetric from Bt rows.  C/D: vgpr r -> row r+8*half.
// ============================================================================
__global__ void wmma_gemm_f32_kernel(const float* __restrict__ A,
                                     const float* __restrict__ Bt,
                                     const float* __restrict__ bias,
                                     float* __restrict__ D,
                                     int tilesN, int K, int ldd) {
  const int tile = blockIdx.x;
  const int tm = tile / tilesN;
  const int tn = tile % tilesN;
  const int lane = threadIdx.x;       // blockDim.x == 32, EXEC all ones
  const int half = lane >> 4;
  const int m16  = lane & 15;
  const float* arow = A  + (size_t)(tm * 16 + m16) * K + 2 * half;
  const float* brow = Bt + (size_t)(tn * 16 + m16) * K + 2 * half;
  v8f c = {0.f, 0.f, 0.f, 0.f, 0.f, 0.f, 0.f, 0.f};
  for (int k0 = 0; k0 < K; k0 += 4) {
    const v2f a = *(const v2f*)(arow + k0);
    const v2f b = *(const v2f*)(brow + k0);
    // (neg_a, A, neg_b, B, c_mod, C, reuse_a, reuse_b)
    c = __builtin_amdgcn_wmma_f32_16x16x4_f32(false, a, false, b, (short)0, c,
                                              false, false);
  }
  const float bb = bias ? bias[tn * 16 + m16] : 0.0f;
  for (int r = 0; r < 8; ++r) {
    D[(size_t)(tm * 16 + r + 8 * half) * ldd + tn * 16 + m16] = c[r] + bb;
  }
}

// ============================================================================
// Kernel 5: per-row top-5 of similarity (256 x 32), one thread per row.
// Ties resolve to lowest index (strict '>' scan), matching jax.lax.top_k.
// reduce_sim = sum(top-5 values over all rows) / B  (since
// key_norm[idx]·xnorm == similarity[., idx]).  Single block -> no atomics.
// ============================================================================
__global__ void topk_kernel(const float* __restrict__ sim,
                            float* __restrict__ idx_f,       // (B,5) as float
                            int* __restrict__ idx_i,         // (B,5) ints (ws)
                            float* __restrict__ reduce_sim) {
  const int b = threadIdx.x;           // 256
  float vals[P_];
  const float* s = sim + (size_t)b * P_;
  for (int i = 0; i < P_; ++i) vals[i] = s[i];
  float mysum = 0.0f;
  for (int k = 0; k < TOPK; ++k) {
    int best = 0; float bv = vals[0];
    for (int i = 1; i < P_; ++i)
      if (vals[i] > bv) { bv = vals[i]; best = i; }
    idx_f[(size_t)b * TOPK + k] = (float)best;
    idx_i[(size_t)b * TOPK + k] = best;
    mysum += bv;
    vals[best] = -INFINITY;
  }
  __shared__ float red[B_];
  red[b] = mysum;
  __syncthreads();
  for (int s2 = 128; s2 > 0; s2 >>= 1) {
    if (b < s2) red[b] += red[b + s2];
    __syncthreads();
  }
  if (b == 0) *reduce_sim = red[0] / (float)B_;
}

// ============================================================================
// Kernel 6: gather prompt_embed rows into out[b, 0:20, :].
// block = (kl, b); 192 threads x float4 = one 768-wide row.
// ============================================================================
__global__ void gather_kernel(const float* __restrict__ pe,   // (128,768)
                              const int* __restrict__ idx,    // (B,5)
                              float* __restrict__ out) {      // (B,216,768)
  const int kl = blockIdx.x;          // 0..19
  const int b  = blockIdx.y;          // 0..255
  const int k = kl >> 2, l = kl & 3;
  const int p = idx[(size_t)b * TOPK + k];
  const float4* src = (const float4*)(pe + (size_t)(p * 4 + l) * D_);
  float4* dst = (float4*)(out + (size_t)b * LSEQ * D_ + (size_t)kl * D_);
  dst[threadIdx.x] = src[threadIdx.x];
}

// ============================================================================
extern "C" void kernel_launch(void* const* d_in, const int* in_sizes, int n_in,
                              void* d_out, int out_size, void* d_ws, size_t ws_size,
                              hipStream_t stream) {
  const float* x_embed    = (const float*)d_in[0];  // (256,196,768)
  const float* prompt     = (const float*)d_in[1];  // (32,3,32,32)
  const float* prompt_key = (const float*)d_in[2];  // (32,768)
  const float* conv_w     = (const float*)d_in[3];  // (768,3,16,16) == Bt(768x768)
  const float* conv_b     = (const float*)d_in[4];  // (768,)

  float* out = (float*)d_out;
  // output tuple layout (flat, return order)
  const size_t OUT_RS  = (size_t)B_ * LSEQ * D_;    // reduce_sim scalar
  const size_t OUT_SIM = OUT_RS + 1;                // similarity (256,32)
  const size_t OUT_IDX = OUT_SIM + (size_t)B_ * P_; // idx (256,5) as float

  // workspace layout (floats), ~1.7 MB total
  float* ws      = (float*)d_ws;
  float* w_xnorm = ws;                                  // 256*768
  float* w_pnorm = w_xnorm + (size_t)B_ * D_;           // 32*768
  float* w_A     = w_pnorm + (size_t)P_ * D_;           // 128*768
  float* w_pe    = w_A + (size_t)MROWS * D_;            // 128*768
  int*   w_idx   = (int*)(w_pe + (size_t)MROWS * D_);   // 256*5 ints

  // 1) copy x_embed into concat slot + mean + normalize (dominant HBM pass)
  mean_norm_copy_kernel<<<B_, 192, 0, stream>>>(x_embed, out, w_xnorm);
  // 2) prompt_key row normalize
  rownorm_kernel<<<P_, 192, 0, stream>>>(prompt_key, w_pnorm);
  // 3) extract conv patches into GEMM A matrix
  patchify_kernel<<<MROWS, 256, 0, stream>>>(prompt, w_A);
  // 4) prompt_embed = A * conv_w^T + conv_b   (128x768x768, f32 WMMA)
  wmma_gemm_f32_kernel<<<(MROWS / 16) * (D_ / 16), 32, 0, stream>>>(
      w_A, conv_w, conv_b, w_pe, D_ / 16, D_, D_);
  // 5) similarity = xnorm * pnorm^T  (256x32x768, f32 WMMA) -> out directly
  wmma_gemm_f32_kernel<<<(B_ / 16) * (P_ / 16), 32, 0, stream>>>(
      w_xnorm, w_pnorm, nullptr, out + OUT_SIM, P_ / 16, D_, P_);
  // 6) top-5 per row, idx (float to out, int to ws), reduce_sim
  topk_kernel<<<1, B_, 0, stream>>>(out + OUT_SIM, out + OUT_IDX, w_idx,
                                    out + OUT_RS);
  // 7) gather selected prompt_embed rows into out[b, 0:20, :]
  dim3 ggrid(TOPK * 4, B_);
  gather_kernel<<<ggrid, 192, 0, stream>>>(w_pe, w_idx, out);
}